// DependencyAnalyzer_90580860273223
// MI455X (gfx1250) — compile-verified
//
#include <hip/hip_runtime.h>

typedef __attribute__((ext_vector_type(2))) float v2f;
typedef __attribute__((ext_vector_type(8))) float v8f;

#define N_NODES 8192
#define F_IN    10
#define H_DIM   64
#define N_EDGES 524288

// ---------------------------------------------------------------------------
// h0 = relu(features @ W_node + b_node)   [8192 x 10] @ [10 x 64]
// one thread per (node, col)
// ---------------------------------------------------------------------------
__global__ void embed_kernel(const float* __restrict__ feat,
                             const float* __restrict__ Wn,
                             const float* __restrict__ bn,
                             float* __restrict__ h) {
    int idx = blockIdx.x * blockDim.x + threadIdx.x;   // node*64 + col
    int n = idx >> 6;
    int c = idx & 63;
    float acc = bn[c];
#pragma unroll
    for (int f = 0; f < F_IN; ++f)
        acc = fmaf(feat[n * F_IN + f], Wn[f * H_DIM + c], acc);
    h[idx] = fmaxf(acc, 0.0f);
}

__global__ void zero_kernel(float* __restrict__ p) {
    p[blockIdx.x * blockDim.x + threadIdx.x] = 0.0f;
}

// ---------------------------------------------------------------------------
// agg[dst] += h[src]  over all edges; 64 lanes per edge (coalesced columns)
// ---------------------------------------------------------------------------
__global__ void scatter_kernel(const float* __restrict__ h,
                               const int* __restrict__ edges,
                               float* __restrict__ agg) {
    int t = blockIdx.x * blockDim.x + threadIdx.x;
    int e = t >> 6;
    int c = t & 63;
    int src = edges[e * 2 + 0];
    int dst = edges[e * 2 + 1];
    atomicAdd(&agg[dst * H_DIM + c], h[src * H_DIM + c]);
}

// ---------------------------------------------------------------------------
// mask[n] = (nodes[n] == 2) ? 1 : 0
// ---------------------------------------------------------------------------
__global__ void mask_kernel(const int* __restrict__ nodes,
                            float* __restrict__ m) {
    int i = blockIdx.x * blockDim.x + threadIdx.x;
    m[i] = (nodes[i] == 2) ? 1.0f : 0.0f;
}

// ---------------------------------------------------------------------------
// h_out = relu((h + agg) @ W_conv + b_conv)   [8192x64] @ [64x64]
// One wave per 16x16 output tile, fp32 WMMA 16x16x4, K=64 -> 16 steps.
// wave id: I = wave>>2 (row tile 0..511), J = wave&3 (col tile 0..3)
// ---------------------------------------------------------------------------
__global__ void conv_kernel(const float* __restrict__ h,
                            const float* __restrict__ agg,
                            const float* __restrict__ Wc,
                            const float* __restrict__ bc,
                            float* __restrict__ hout) {
    int wave = (blockIdx.x * blockDim.x + threadIdx.x) >> 5;
    int lane = threadIdx.x & 31;
    int J = wave & 3;
    int I = wave >> 2;
    int half = lane >> 4;      // 0: K sub 0/1, 1: K sub 2/3
    int l = lane & 15;
    int rowA = I * 16 + l;     // A fragment: M = lane%16
    int colB = J * 16 + l;     // B fragment: N = lane%16

    const float* ha = h + (size_t)rowA * H_DIM;
    const float* ga = agg + (size_t)rowA * H_DIM;

    v8f c = {};
#pragma unroll
    for (int kk = 0; kk < 16; ++kk) {
        int kb = kk * 4 + half * 2;
        v2f hv = *(const v2f*)(ha + kb);
        v2f gv = *(const v2f*)(ga + kb);
        v2f a;
        a.x = hv.x + gv.x;
        a.y = hv.y + gv.y;
        v2f b;
        b.x = Wc[(size_t)(kb + 0) * H_DIM + colB];
        b.y = Wc[(size_t)(kb + 1) * H_DIM + colB];
        c = __builtin_amdgcn_wmma_f32_16x16x4_f32(false, a, false, b,
                                                  (short)0, c, false, false);
    }
    float bias = bc[colB];
#pragma unroll
    for (int r = 0; r < 8; ++r) {
        int row = I * 16 + r + half * 8;
        hout[(size_t)row * H_DIM + colB] = fmaxf(c[r] + bias, 0.0f);
    }
}

// ---------------------------------------------------------------------------
// S = h @ h^T; out[0] = S * m_i * m_j (function_deps), out[1] = S (similarity)
// One wave computes a 16x64 strip (4 tiles of 16x16), reusing the A fragment
// 4x per k-step. K=64 -> 16 fp32 WMMA steps per tile.
// wave id: I = wave>>7 (0..511), Jg = wave&127 (group of 4 col tiles)
// ---------------------------------------------------------------------------
__global__ void sim_kernel(const float* __restrict__ h,
                           const float* __restrict__ mask,
                           float* __restrict__ out) {
    int wave = (blockIdx.x * blockDim.x + threadIdx.x) >> 5;
    int lane = threadIdx.x & 31;
    int Jg = wave & 127;
    int I = wave >> 7;
    int half = lane >> 4;
    int l = lane & 15;

    const float* ha  = h + (size_t)(I * 16 + l) * H_DIM;
    const float* hb0 = h + (size_t)(Jg * 64 + 0 * 16 + l) * H_DIM;
    const float* hb1 = h + (size_t)(Jg * 64 + 1 * 16 + l) * H_DIM;
    const float* hb2 = h + (size_t)(Jg * 64 + 2 * 16 + l) * H_DIM;
    const float* hb3 = h + (size_t)(Jg * 64 + 3 * 16 + l) * H_DIM;

    v8f c0 = {}, c1 = {}, c2 = {}, c3 = {};
#pragma unroll
    for (int kk = 0; kk < 16; ++kk) {
        int kb = kk * 4 + half * 2;
        v2f a  = *(const v2f*)(ha + kb);
        v2f b0 = *(const v2f*)(hb0 + kb);
        v2f b1 = *(const v2f*)(hb1 + kb);
        v2f b2 = *(const v2f*)(hb2 + kb);
        v2f b3 = *(const v2f*)(hb3 + kb);
        c0 = __builtin_amdgcn_wmma_f32_16x16x4_f32(false, a, false, b0,
                                                   (short)0, c0, false, false);
        c1 = __builtin_amdgcn_wmma_f32_16x16x4_f32(false, a, false, b1,
                                                   (short)0, c1, false, false);
        c2 = __builtin_amdgcn_wmma_f32_16x16x4_f32(false, a, false, b2,
                                                   (short)0, c2, false, false);
        c3 = __builtin_amdgcn_wmma_f32_16x16x4_f32(false, a, false, b3,
                                                   (short)0, c3, false, false);
    }

    float mrow[8];
#pragma unroll
    for (int r = 0; r < 8; ++r)
        mrow[r] = mask[I * 16 + r + half * 8];

    float* simp = out + (size_t)N_NODES * N_NODES;   // plane 1: similarity
    const v8f* acc[4] = { &c0, &c1, &c2, &c3 };
#pragma unroll
    for (int t = 0; t < 4; ++t) {
        int col = Jg * 64 + t * 16 + l;
        float mcol = mask[col];
        const v8f& c = *acc[t];
#pragma unroll
        for (int r = 0; r < 8; ++r) {
            int row = I * 16 + r + half * 8;
            size_t off = (size_t)row * N_NODES + col;
            float v = c[r];
            simp[off] = v;                      // similarity
            out[off]  = v * mrow[r] * mcol;     // function_deps
        }
    }
}

// ---------------------------------------------------------------------------
extern "C" void kernel_launch(void* const* d_in, const int* in_sizes, int n_in,
                              void* d_out, int out_size, void* d_ws, size_t ws_size,
                              hipStream_t stream) {
    const float* feat  = (const float*)d_in[0];
    const float* Wn    = (const float*)d_in[1];
    const float* bn    = (const float*)d_in[2];
    const float* Wc    = (const float*)d_in[3];
    const float* bc    = (const float*)d_in[4];
    const int*   nodes = (const int*)d_in[5];
    const int*   edges = (const int*)d_in[6];
    float* out = (float*)d_out;

    float* hA  = (float*)d_ws;                 // 8192*64 floats
    float* hB  = hA + (size_t)N_NODES * H_DIM; // 8192*64
    float* agg = hB + (size_t)N_NODES * H_DIM; // 8192*64
    float* msk = agg + (size_t)N_NODES * H_DIM; // 8192

    const int NH = N_NODES * H_DIM;            // 524288

    // h0 = relu(features @ W_node + b_node)
    embed_kernel<<<NH / 256, 256, 0, stream>>>(feat, Wn, bn, hA);

    // two graph-conv rounds
    const float* hin = hA;
    float* hout_arr[2] = { hB, hA };
#pragma unroll
    for (int round = 0; round < 2; ++round) {
        zero_kernel<<<NH / 256, 256, 0, stream>>>(agg);
        scatter_kernel<<<(N_EDGES * 64) / 256, 256, 0, stream>>>(hin, edges, agg);
        // 512 row tiles * 4 col tiles = 2048 waves; 8 waves/block
        conv_kernel<<<2048 * 32 / 256, 256, 0, stream>>>(hin, agg, Wc, bc,
                                                         hout_arr[round]);
        hin = hout_arr[round];
    }
    // final h is in hA after round 1 (hB -> hA)

    mask_kernel<<<N_NODES / 256, 256, 0, stream>>>(nodes, msk);

    // 512 I-tiles * 128 J-groups = 65536 waves; 8 waves/block -> 8192 blocks
    sim_kernel<<<65536 * 32 / 256, 256, 0, stream>>>(hA, msk, out);
}